// YoloKeypointLoss_2336462209777
// MI455X (gfx1250) — compile-verified
//
#include <hip/hip_runtime.h>
#include <math.h>

#define BS      64
#define NUM_KP  17
#define A_DIM   8400
#define CH      (3 * NUM_KP)     // 51 channels
#define GRID_W  80               // IMAGE_SIZE / stride0 = 640/8
#define NROWS   (BS * NUM_KP)    // 1088 (b,k) pairs

typedef __attribute__((ext_vector_type(2))) float v2f;
typedef __attribute__((ext_vector_type(8))) float v8f;

// ---------------------------------------------------------------------------
// Wave-level sum of 32 per-lane partials using V_WMMA_F32_16X16X4_F32.
// A (16x4, lane-striped): lane m (0..15) holds A[m][0..1], lane m+16 holds
// A[m][2..3]. We place each lane's partial in its first slot, zero the second.
// B = all-ones (4x16)  =>  D[m][n] = rowsum_m = x[m] + x[m+16].
// D layout: lanes 0..15 hold D[0..7][n] in 8 VGPRs, lanes 16..31 hold
// D[8..15][n]. Summing a lane's 8 VGPRs gives half the total; shfl_xor(16)
// supplies the other half. EXEC is all ones at the call site (no divergence).
// ---------------------------------------------------------------------------
__device__ __forceinline__ float wave_sum_wmma(float x) {
    v2f a; a[0] = x;    a[1] = 0.0f;
    v2f b; b[0] = 1.0f; b[1] = 1.0f;
    v8f c = {};
    c = __builtin_amdgcn_wmma_f32_16x16x4_f32(
            /*neg_a=*/false, a, /*neg_b=*/false, b,
            /*c_mod=*/(short)0, c, /*reuse_a=*/false, /*reuse_b=*/false);
    float s = c[0] + c[1] + c[2] + c[3] + c[4] + c[5] + c[6] + c[7];
    s += __shfl_xor(s, 16, 32);   // combine the two row-halves -> full wave sum
    return s;
}

// ---------------------------------------------------------------------------
// Kernel A: per-(b,k) row reduction of max(log1p(-conf), -100) over 8400
// elements. One block per row; float4 streaming loads; WMMA wave reduce;
// block partial written (f64) to workspace. Fully deterministic.
// ---------------------------------------------------------------------------
__global__ void __launch_bounds__(256)
conf_row_sum_kernel(const float* __restrict__ output,
                    double* __restrict__ conf_part) {
    __shared__ float wsum[8];
    const int r = blockIdx.x;             // 0..NROWS-1
    const int b = r / NUM_KP;
    const int k = r - b * NUM_KP;
    const float4* row = (const float4*)(output + ((size_t)b * CH + 3 * k + 2) * A_DIM);

    float local = 0.0f;
    for (int i = threadIdx.x; i < A_DIM / 4; i += 256) {
        float4 v = row[i];
        local += fmaxf(log1pf(-v.x), -100.0f)
               + fmaxf(log1pf(-v.y), -100.0f)
               + fmaxf(log1pf(-v.z), -100.0f)
               + fmaxf(log1pf(-v.w), -100.0f);
    }

    float s = wave_sum_wmma(local);                 // all 256 lanes active
    const int wave = threadIdx.x >> 5;
    const int lane = threadIdx.x & 31;
    if (lane == 0) wsum[wave] = s;
    __syncthreads();
    if (threadIdx.x == 0) {
        double blk = 0.0;
        for (int w = 0; w < 8; ++w) blk += (double)wsum[w];
        conf_part[r] = blk;
    }
}

// ---------------------------------------------------------------------------
// Kernel B: sparse corrections, one thread per (b,k). Every slot written each
// launch (no stale workspace). Matches reference index math exactly.
// ---------------------------------------------------------------------------
__global__ void __launch_bounds__(256)
corrections_kernel(const float* __restrict__ output,
                   const float* __restrict__ gt,       // (BS, NUM_KP, 2)
                   const int*   __restrict__ vis,      // (BS, NUM_KP)
                   float* __restrict__ cl,             // logp - log1mp
                   float* __restrict__ cx,             // (xg-gx)^2
                   float* __restrict__ cy) {           // (yg-gy)^2
    const int t = blockIdx.x * blockDim.x + threadIdx.x;
    if (t >= NROWS) return;
    const int b = t / NUM_KP;
    const int k = t - b * NUM_KP;

    const float gx = gt[(size_t)t * 2 + 0];
    const float gy = gt[(size_t)t * 2 + 1];
    const int   v  = vis[t];

    float rl = 0.0f, rx = 0.0f, ry = 0.0f;
    if (v == 1) {
        const float bx = floorf(gx * 0.125f);       // /8 exact in f32
        const float by = floorf(gy * 0.125f);
        const int idx  = (int)(by * (float)GRID_W + bx);   // <= 6399, exact
        const size_t base = ((size_t)b * CH + 3 * k) * A_DIM + (size_t)idx;
        const float xg = output[base];
        const float yg = output[base + A_DIM];
        const float cg = output[base + 2 * (size_t)A_DIM];
        const float dx = xg - gx, dy = yg - gy;
        const float logp   = fmaxf(logf(cg),    -100.0f);
        const float log1mp = fmaxf(log1pf(-cg), -100.0f);
        rl = logp - log1mp;
        rx = dx * dx;
        ry = dy * dy;
    }
    cl[t] = rl; cx[t] = rx; cy[t] = ry;
}

// ---------------------------------------------------------------------------
// Kernel C: deterministic final combine. Single block, fixed-order strided
// accumulation + fixed-order LDS tree in f64.
// ---------------------------------------------------------------------------
__global__ void __launch_bounds__(256)
finalize_kernel(const double* __restrict__ conf_part,
                const float*  __restrict__ cl,
                const float*  __restrict__ cx,
                const float*  __restrict__ cy,
                float* __restrict__ out) {
    __shared__ double s1[256];
    __shared__ double s2[256];
    double q1 = 0.0, q2 = 0.0;
    for (int i = threadIdx.x; i < NROWS; i += 256) {
        q1 += conf_part[i] + (double)cl[i];          // S_all + S_corr
        q2 += (double)cx[i] + (double)cy[i];         // x_loss + y_loss sums
    }
    s1[threadIdx.x] = q1;
    s2[threadIdx.x] = q2;
    __syncthreads();
    for (int off = 128; off > 0; off >>= 1) {
        if (threadIdx.x < (unsigned)off) {
            s1[threadIdx.x] += s1[threadIdx.x + off];
            s2[threadIdx.x] += s2[threadIdx.x + off];
        }
        __syncthreads();
    }
    if (threadIdx.x == 0) {
        const double n_conf = (double)BS * (double)NUM_KP * (double)A_DIM;
        const double conf_loss = -s1[0] / n_conf;
        out[0] = (float)(conf_loss + s2[0] / (double)BS);
    }
}

// ---------------------------------------------------------------------------
extern "C" void kernel_launch(void* const* d_in, const int* in_sizes, int n_in,
                              void* d_out, int out_size, void* d_ws, size_t ws_size,
                              hipStream_t stream) {
    const float* output = (const float*)d_in[0];   // (64, 51, 8400) f32
    // d_in[1] = target (unused by the loss value)
    const float* gt  = (const float*)d_in[2];      // (64, 17, 2) f32
    const int*   vis = (const int*)d_in[3];        // (64, 17) i32

    // Workspace layout: 1088 f64 block partials + 3 x 1088 f32 corrections.
    double* conf_part = (double*)d_ws;
    float*  cl = (float*)(conf_part + NROWS);
    float*  cx = cl + NROWS;
    float*  cy = cx + NROWS;

    conf_row_sum_kernel<<<NROWS, 256, 0, stream>>>(output, conf_part);
    corrections_kernel<<<(NROWS + 255) / 256, 256, 0, stream>>>(output, gt, vis, cl, cx, cy);
    finalize_kernel<<<1, 256, 0, stream>>>(conf_part, cl, cx, cy, (float*)d_out);
    (void)in_sizes; (void)n_in; (void)out_size; (void)ws_size;
}